// GNN_12051678233252
// MI455X (gfx1250) — compile-verified
//
#include <hip/hip_runtime.h>
#include <hip/hip_bf16.h>

#define NN     100000
#define EE     1600000
#define ET     1700000   // EE + NN self loops
#define CC     128
#define DIN    6
#define OUTD   2

typedef __attribute__((ext_vector_type(2))) float v2f;
typedef __attribute__((ext_vector_type(4))) float v4f;
typedef __attribute__((ext_vector_type(8))) float v8f;

// ---------- helpers ----------
__device__ __forceinline__ float waveReduceSum(float v) {
  #pragma unroll
  for (int off = 16; off > 0; off >>= 1) v += __shfl_xor(v, off, 32);
  return v;
}
// order-preserving float->uint for atomicMax-based segment max
__device__ __forceinline__ unsigned int orderEnc(float f) {
  unsigned int u = __float_as_uint(f);
  return (u & 0x80000000u) ? ~u : (u | 0x80000000u);
}
__device__ __forceinline__ float orderDec(unsigned int u) {
  unsigned int b = (u & 0x80000000u) ? (u & 0x7FFFFFFFu) : ~u;
  return __uint_as_float(b);
}

// ---------- 0: init accumulators ----------
__global__ void k_init(float* __restrict__ h, float* __restrict__ den,
                       float* __restrict__ deg, unsigned int* __restrict__ lmaxb) {
  size_t i = (size_t)blockIdx.x * blockDim.x + threadIdx.x;
  if (i < (size_t)NN * CC) h[i] = 0.f;
  if (i < NN) { den[i] = 0.f; deg[i] = 0.f; lmaxb[i] = 0u; }  // 0 encodes < all reals
}

// ---------- 1: xl = x@Wl+bl, xr = x@Wr+br  (N x 6 -> N x 128) ----------
__global__ void k_proj(const float* __restrict__ x,
                       const float* __restrict__ Wl, const float* __restrict__ bl,
                       const float* __restrict__ Wr, const float* __restrict__ br,
                       float* __restrict__ xl, float* __restrict__ xr) {
  int n = blockIdx.x;           // one block (128 threads) per node
  int c = threadIdx.x;
  __shared__ float xs[8];
  if (c < DIN) xs[c] = x[(size_t)n * DIN + c];
  __syncthreads();
  float sl = bl[c], sr = br[c];
  #pragma unroll
  for (int k = 0; k < DIN; ++k) {
    sl = fmaf(xs[k], Wl[k * CC + c], sl);
    sr = fmaf(xs[k], Wr[k * CC + c], sr);
  }
  xl[(size_t)n * CC + c] = sl;
  xr[(size_t)n * CC + c] = sr;
}

// ---------- 2: per-edge GATv2 logits + segment max (wave per edge) ----------
__global__ void k_edge_logits(const int* __restrict__ ei, const float* __restrict__ ew,
                              const float* __restrict__ xl, const float* __restrict__ xr,
                              const float* __restrict__ We, const float* __restrict__ att,
                              float* __restrict__ logits, unsigned int* __restrict__ lmaxb) {
  int e = blockIdx.x * (blockDim.x >> 5) + (threadIdx.x >> 5);
  if (e >= ET) return;
  int lane = threadIdx.x & 31;
  int s, d; float ea0, ea1;
  if (e < EE) { s = ei[e]; d = ei[EE + e]; ea0 = ew[2 * e]; ea1 = ew[2 * e + 1]; }
  else        { s = d = e - EE; ea0 = 0.f; ea1 = 0.f; }
  int c0 = lane * 4;
  const v4f a = *(const v4f*)(xl + (size_t)s * CC + c0);
  const v4f b = *(const v4f*)(xr + (size_t)d * CC + c0);
  float t = 0.f;
  #pragma unroll
  for (int j = 0; j < 4; ++j) {
    int c = c0 + j;
    float m = a[j] + b[j] + ea0 * We[c] + ea1 * We[CC + c];
    m = (m > 0.f) ? m : 0.2f * m;              // leaky_relu(0.2)
    t = fmaf(m, att[c], t);
  }
  t = waveReduceSum(t);
  if (lane == 0) {
    logits[e] = t;
    atomicMax(&lmaxb[d], orderEnc(t));
  }
}

// ---------- 3: ex = exp(logit - max), den = segsum(ex) ----------
__global__ void k_exp(const int* __restrict__ ei, const float* __restrict__ logits,
                      const unsigned int* __restrict__ lmaxb,
                      float* __restrict__ alpha, float* __restrict__ den) {
  int e = blockIdx.x * blockDim.x + threadIdx.x;
  if (e >= ET) return;
  int d = (e < EE) ? ei[EE + e] : (e - EE);
  float ex = expf(logits[e] - orderDec(lmaxb[d]));
  alpha[e] = ex;                                 // staged in alpha output slot
  atomicAdd(&den[d], ex);
}

// ---------- 4: alpha = ex/den; deg += alpha; h[dst] += alpha * xl[src] ----------
__global__ void k_alpha_agg(const int* __restrict__ ei, const float* __restrict__ xl,
                            const float* __restrict__ den, float* __restrict__ alpha,
                            float* __restrict__ deg, float* __restrict__ h) {
  int e = blockIdx.x * (blockDim.x >> 5) + (threadIdx.x >> 5);
  if (e >= ET) return;
  int lane = threadIdx.x & 31;
  int s, d;
  if (e < EE) { s = ei[e]; d = ei[EE + e]; } else { s = d = e - EE; }
  float a = 0.f;
  if (lane == 0) {
    a = alpha[e] / den[d];
    alpha[e] = a;
    atomicAdd(&deg[d], a);
  }
  a = __shfl(a, 0, 32);
  int c0 = lane * 4;
  const v4f xv = *(const v4f*)(xl + (size_t)s * CC + c0);
  float* hp = h + (size_t)d * CC + c0;
  #pragma unroll
  for (int j = 0; j < 4; ++j) atomicAdd(hp + j, a * xv[j]);
}

// ---------- 5: h = relu(h + b_gat); h2 = 0; dinv = rsqrt(deg) ----------
__global__ void k_finalize_h(float* __restrict__ h, const float* __restrict__ b_gat,
                             float* __restrict__ h2, const float* __restrict__ deg,
                             float* __restrict__ dinv) {
  size_t i = (size_t)blockIdx.x * blockDim.x + threadIdx.x;
  if (i >= (size_t)NN * CC) return;
  float v = h[i] + b_gat[i & (CC - 1)];
  h[i] = fmaxf(v, 0.f);
  h2[i] = 0.f;
  if ((i & (CC - 1)) == 0) {
    int n = (int)(i >> 7);
    float dg = deg[n];
    dinv[n] = (dg > 0.f) ? rsqrtf(dg) : 0.f;
  }
}

// ---------- 6: hg = h @ W2  (f32 WMMA 16x16x4, LDS-staged A tile) ----------
#define SH_LD 132   // 128 + 4 pad: A-frag reads hit distinct LDS banks
__global__ void k_gemm_hW2(const float* __restrict__ h, const float* __restrict__ W2,
                           float* __restrict__ hg) {
  __shared__ float sh[16 * SH_LD];
  const int rowbase = blockIdx.x * 16;
  const int tid = threadIdx.x;
  // coalesced 16x128 tile load
  for (int i = tid; i < 16 * CC; i += 256) {
    int r = i >> 7, c = i & (CC - 1);
    sh[r * SH_LD + c] = h[(size_t)(rowbase + r) * CC + c];
  }
  __syncthreads();
  const int wave = tid >> 5, lane = tid & 31;
  const int colbase = wave * 16;         // 8 waves cover the 128 output cols
  const int m = lane & 15, half = lane >> 4;
  v8f acc = {};
  #pragma unroll 4
  for (int k0 = 0; k0 < CC; k0 += 4) {
    // A(16x4) frag: lane l holds A[l%16][2*(l/16)+v]  (ISA 7.12.2, 32-bit A 16x4)
    const int ka = k0 + 2 * half;
    v2f a; a.x = sh[m * SH_LD + ka]; a.y = sh[m * SH_LD + ka + 1];
    // B(4x16) frag: VGPR v, lane l holds B[2*(l/16)+v][l%16] (row-striped)
    v2f b; b.x = W2[(size_t)ka * CC + colbase + m];
           b.y = W2[(size_t)(ka + 1) * CC + colbase + m];
    acc = __builtin_amdgcn_wmma_f32_16x16x4_f32(
        /*neg_a=*/false, a, /*neg_b=*/false, b,
        /*c_mod=*/(short)0, acc, /*reuse_a=*/false, /*reuse_b=*/false);
  }
  // D(16x16) f32: VGPR i, lanes 0-15 -> M=i, lanes 16-31 -> M=i+8; N = lane%16
  #pragma unroll
  for (int i = 0; i < 8; ++i) {
    int row = rowbase + i + 8 * half;
    hg[(size_t)row * CC + colbase + m] = acc[i];
  }
}

// ---------- 7: h2[dst] += dinv[src]*alpha*dinv[dst] * hg[src] ----------
__global__ void k_gcn_agg(const int* __restrict__ ei, const float* __restrict__ hg,
                          const float* __restrict__ alpha, const float* __restrict__ dinv,
                          float* __restrict__ h2) {
  int e = blockIdx.x * (blockDim.x >> 5) + (threadIdx.x >> 5);
  if (e >= ET) return;
  int lane = threadIdx.x & 31;
  int s, d;
  if (e < EE) { s = ei[e]; d = ei[EE + e]; } else { s = d = e - EE; }
  float nrm = 0.f;
  if (lane == 0) nrm = dinv[s] * alpha[e] * dinv[d];
  nrm = __shfl(nrm, 0, 32);
  int c0 = lane * 4;
  const v4f g = *(const v4f*)(hg + (size_t)s * CC + c0);
  float* hp = h2 + (size_t)d * CC + c0;
  #pragma unroll
  for (int j = 0; j < 4; ++j) atomicAdd(hp + j, nrm * g[j]);
}

// ---------- 8: out = relu(h2 + b_gcn) @ W3 + b3  (wave per node) ----------
__global__ void k_final(const float* __restrict__ h2, const float* __restrict__ b_gcn,
                        const float* __restrict__ W3, const float* __restrict__ b3,
                        float* __restrict__ out) {
  int n = blockIdx.x * (blockDim.x >> 5) + (threadIdx.x >> 5);
  if (n >= NN) return;
  int lane = threadIdx.x & 31;
  float s0 = 0.f, s1 = 0.f;
  #pragma unroll
  for (int j = 0; j < 4; ++j) {
    int c = lane + 32 * j;
    float v = fmaxf(h2[(size_t)n * CC + c] + b_gcn[c], 0.f);
    s0 = fmaf(v, W3[c * OUTD + 0], s0);
    s1 = fmaf(v, W3[c * OUTD + 1], s1);
  }
  s0 = waveReduceSum(s0);
  s1 = waveReduceSum(s1);
  if (lane == 0) {
    out[(size_t)n * OUTD + 0] = s0 + b3[0];
    out[(size_t)n * OUTD + 1] = s1 + b3[1];
  }
}

// ---------- 9: emit stack([src,dst]) with self loops appended ----------
__global__ void k_edges_out(const int* __restrict__ ei, int* __restrict__ outSrc,
                            int* __restrict__ outDst) {
  int e = blockIdx.x * blockDim.x + threadIdx.x;
  if (e >= ET) return;
  int s, d;
  if (e < EE) { s = ei[e]; d = ei[EE + e]; } else { s = d = e - EE; }
  outSrc[e] = s;
  outDst[e] = d;
}

extern "C" void kernel_launch(void* const* d_in, const int* in_sizes, int n_in,
                              void* d_out, int out_size, void* d_ws, size_t ws_size,
                              hipStream_t stream) {
  const float* x    = (const float*)d_in[0];
  const int*   ei   = (const int*)  d_in[1];
  const float* ew   = (const float*)d_in[2];
  const float* Wl   = (const float*)d_in[3];
  const float* bl   = (const float*)d_in[4];
  const float* Wr   = (const float*)d_in[5];
  const float* br   = (const float*)d_in[6];
  const float* We   = (const float*)d_in[7];
  const float* att  = (const float*)d_in[8];
  const float* bgat = (const float*)d_in[9];
  const float* W2   = (const float*)d_in[10];
  const float* bgcn = (const float*)d_in[11];
  const float* W3   = (const float*)d_in[12];
  const float* b3   = (const float*)d_in[13];

  // outputs: [N*OUT floats][src Et ints][dst Et ints][alpha Et floats]
  float* out       = (float*)d_out;
  int*   eidx_out  = (int*)d_out + (size_t)NN * OUTD;
  float* alpha_out = (float*)d_out + (size_t)NN * OUTD + 2 * (size_t)ET;

  // workspace: buf0 (xl -> h2), buf1 (xr -> hg), buf2 (h), logits, per-node scalars
  const size_t NC = (size_t)NN * CC;
  float*        buf0   = (float*)d_ws;        // xl, later h2
  float*        buf1   = buf0 + NC;           // xr, later hg
  float*        buf2   = buf1 + NC;           // h
  float*        logits = buf2 + NC;           // Et
  unsigned int* lmaxb  = (unsigned int*)(logits + ET);
  float*        den    = (float*)(lmaxb + NN);
  float*        deg    = den + NN;
  float*        dinv   = deg + NN;

  const int edgeWaveBlocks = (ET + 7) / 8;    // 8 waves (edges) per 256-thread block
  const int nodeWaveBlocks = (NN + 7) / 8;

  k_init       <<<(int)((NC + 255) / 256), 256, 0, stream>>>(buf2, den, deg, lmaxb);
  k_proj       <<<NN, CC, 0, stream>>>(x, Wl, bl, Wr, br, buf0, buf1);
  k_edge_logits<<<edgeWaveBlocks, 256, 0, stream>>>(ei, ew, buf0, buf1, We, att, logits, lmaxb);
  k_exp        <<<(ET + 255) / 256, 256, 0, stream>>>(ei, logits, lmaxb, alpha_out, den);
  k_alpha_agg  <<<edgeWaveBlocks, 256, 0, stream>>>(ei, buf0, den, alpha_out, deg, buf2);
  k_finalize_h <<<(int)((NC + 255) / 256), 256, 0, stream>>>(buf2, bgat, buf0, deg, dinv);
  k_gemm_hW2   <<<NN / 16, 256, 0, stream>>>(buf2, W2, buf1);   // N=100000 = 16*6250
  k_gcn_agg    <<<edgeWaveBlocks, 256, 0, stream>>>(ei, buf1, alpha_out, dinv, buf0);
  k_final      <<<nodeWaveBlocks, 256, 0, stream>>>(buf0, bgcn, W3, b3, out);
  k_edges_out  <<<(ET + 255) / 256, 256, 0, stream>>>(ei, eidx_out, eidx_out + ET);
}